// EmbeddingMul_73564199845928
// MI455X (gfx1250) — compile-verified
//
#include <hip/hip_runtime.h>

// Embedding gather: out[row, :] = weight[idx[row], :]
//   idx:    16384 int   (BPTT=2048 x BSIZE=8, flattened)
//   weight: 50257 x 1024 float
//   out:    16384 x 1024 float
//
// Pure bandwidth problem (~128 MiB traffic -> ~5.5us at 23.3 TB/s).
// Fast path: batch 8 rows per block -- 8 scalar index loads (one s_clause),
// then 8 global_load_b128 in flight per wave, then 8 NT global_store_b128.
// NT stores keep the write-once output from evicting the ~196MB weight
// table out of the 192MB L2. global_prefetch_b8 warms L2 for the next block.

typedef float f32x4 __attribute__((ext_vector_type(4)));

#define EMB            1024
#define VEC            4
#define THREADS        (EMB / VEC)   /* 256 threads = 8 wave32: one 4KB row/iter */
#define ROWS_PER_BLOCK 8

__global__ __launch_bounds__(THREADS)
void embedding_gather_kernel(const int* __restrict__ idx,
                             const float* __restrict__ weight,
                             float* __restrict__ out,
                             int nrows)
{
    const int t    = threadIdx.x;
    const int row0 = blockIdx.x * ROWS_PER_BLOCK;
    const long long col = (long long)t * VEC;

    if (row0 + ROWS_PER_BLOCK <= nrows) {
        // ---- fast path: fully pipelined, 8 b128 loads in flight ----

        // 1) all 8 block-uniform indices (scalar loads, clause-able)
        long long woff[ROWS_PER_BLOCK];
#pragma unroll
        for (int j = 0; j < ROWS_PER_BLOCK; ++j)
            woff[j] = (long long)idx[row0 + j] * EMB;

        // Warm L2 for the next block's first row while we work
        // (global_prefetch_b8, gfx1250 path).
        if (row0 + ROWS_PER_BLOCK < nrows) {
            const long long poff = (long long)idx[row0 + ROWS_PER_BLOCK] * EMB;
            __builtin_prefetch(weight + poff + col, /*rw=*/0, /*loc=*/1);
        }

        // 2) issue all gathers back-to-back (8 x global_load_b128 per lane)
        f32x4 v[ROWS_PER_BLOCK];
#pragma unroll
        for (int j = 0; j < ROWS_PER_BLOCK; ++j)
            v[j] = *(const f32x4*)(weight + woff[j] + col);

        // 3) drain with non-temporal stores; store j only waits on load j
        //    (loads complete in order -> graduated s_wait_loadcnt)
#pragma unroll
        for (int j = 0; j < ROWS_PER_BLOCK; ++j) {
            f32x4* dst = (f32x4*)(out + (long long)(row0 + j) * EMB + col);
            __builtin_nontemporal_store(v[j], dst);
        }
    } else {
        // ---- tail path: per-row guard (nrows not a multiple of 8) ----
        for (int j = 0; j < ROWS_PER_BLOCK; ++j) {
            const int row = row0 + j;
            if (row >= nrows) return;
            const long long woff = (long long)idx[row] * EMB;
            const f32x4 v = *(const f32x4*)(weight + woff + col);
            f32x4* dst = (f32x4*)(out + (long long)row * EMB + col);
            __builtin_nontemporal_store(v, dst);
        }
    }
}

extern "C" void kernel_launch(void* const* d_in, const int* in_sizes, int n_in,
                              void* d_out, int out_size, void* d_ws, size_t ws_size,
                              hipStream_t stream)
{
    const int*   idx    = (const int*)d_in[0];    // input: (BPTT, BSIZE) int, flat
    const float* weight = (const float*)d_in[1];  // (VOCAB, EMSIZE) f32
    float*       out    = (float*)d_out;          // (BPTT, BSIZE, EMSIZE) f32

    const int nrows = in_sizes[0];                // 16384
    const int grid  = (nrows + ROWS_PER_BLOCK - 1) / ROWS_PER_BLOCK;

    embedding_gather_kernel<<<grid, THREADS, 0, stream>>>(idx, weight, out, nrows);
}